// SurfaceAbstractionCD_26104811225239
// MI455X (gfx1250) — compile-verified
//
#include <hip/hip_runtime.h>
#include <math.h>

// ---------------------------------------------------------------------------
// Problem constants (match reference)
// ---------------------------------------------------------------------------
constexpr int Bb = 4;
constexpr int Nn = 16384;
constexpr int Ss = 1024;
constexpr int Kk = 32;
constexpr int CF = 64;                  // C_FEAT
constexpr int P  = Bb * Ss * Kk;        // 131072 positions (B,S,K flattened)
constexpr int PS = Bb * Ss;             // 4096 pooled positions
constexpr float BN_EPS = 1e-5f;
constexpr float PI_F = 3.14159265358979323846f;

typedef __attribute__((ext_vector_type(2))) float v2f;
typedef __attribute__((ext_vector_type(8))) float v8f;

// CDNA5 f32 WMMA: D(16x16) = A(16x4) * B(4x16) + C
__device__ __forceinline__ v8f wmma4(v2f a, v2f b, v8f c) {
  return __builtin_amdgcn_wmma_f32_16x16x4_f32(false, a, false, b, (short)0, c,
                                               false, false);
}

// ---------------------------------------------------------------------------
// 1. Farthest point sampling. One block (1024 threads) per batch.
//    Each thread owns 16 points kept in registers; argmax via LDS tree.
//    Also emits new_center / new_normal (outputs 0 and 1).
// ---------------------------------------------------------------------------
__global__ void fps_kernel(const float* __restrict__ center,
                           const float* __restrict__ normal,
                           int* __restrict__ fps_idx,
                           float* __restrict__ out_center,
                           float* __restrict__ out_normal) {
  const int b   = blockIdx.x;
  const int tid = threadIdx.x;
  const int n0  = tid * 16;
  const size_t base = (size_t)b * Nn;

  float px[16], py[16], pz[16], dist[16];
#pragma unroll
  for (int j = 0; j < 16; ++j) {
    size_t o = (base + n0 + j) * 3;
    px[j] = center[o + 0];
    py[j] = center[o + 1];
    pz[j] = center[o + 2];
    dist[j] = 1e10f;
  }

  __shared__ float rd[1024];
  __shared__ int   ri[1024];
  __shared__ int   cur;
  if (tid == 0) cur = 0;
  __syncthreads();

  for (int it = 0; it < Ss; ++it) {
    const int far = cur;
    if (tid == 0) {
      fps_idx[b * Ss + it] = far;
      size_t go = (base + far) * 3;
      size_t oo = ((size_t)b * Ss + it) * 3;
      out_center[oo + 0] = center[go + 0];
      out_center[oo + 1] = center[go + 1];
      out_center[oo + 2] = center[go + 2];
      out_normal[oo + 0] = normal[go + 0];
      out_normal[oo + 1] = normal[go + 1];
      out_normal[oo + 2] = normal[go + 2];
    }
    const float cx = center[(base + far) * 3 + 0];
    const float cy = center[(base + far) * 3 + 1];
    const float cz = center[(base + far) * 3 + 2];

    float bd = -1.0f;
    int   bi = n0;
#pragma unroll
    for (int j = 0; j < 16; ++j) {
      float dx = px[j] - cx, dy = py[j] - cy, dz = pz[j] - cz;
      float d = dx * dx + dy * dy + dz * dz;
      dist[j] = fminf(dist[j], d);
      if (dist[j] > bd) { bd = dist[j]; bi = n0 + j; }
    }
    rd[tid] = bd;
    ri[tid] = bi;
    __syncthreads();
    for (int o = 512; o > 0; o >>= 1) {
      if (tid < o) {
        float od = rd[tid + o];
        int   oi = ri[tid + o];
        if (od > rd[tid] || (od == rd[tid] && oi < ri[tid])) {
          rd[tid] = od;
          ri[tid] = oi;
        }
      }
      __syncthreads();
    }
    if (tid == 0) cur = ri[0];
    __syncthreads();
  }
}

// ---------------------------------------------------------------------------
// 2. KNN: one block per (b,s). Full distance array in 64KB LDS, 32 argmin
//    passes with lowest-index tie-breaking.
// ---------------------------------------------------------------------------
__global__ void knn_kernel(const float* __restrict__ center,
                           const int* __restrict__ fps_idx,
                           int* __restrict__ knn_idx) {
  constexpr int NT = 256;
  __shared__ float sd[Nn];            // 64 KB
  __shared__ float rd[NT];
  __shared__ int   ri[NT];

  const int s = blockIdx.x;
  const int b = blockIdx.y;
  const int tid = threadIdx.x;
  const size_t base = (size_t)b * Nn;

  const int fi = fps_idx[b * Ss + s];
  const float qx = center[(base + fi) * 3 + 0];
  const float qy = center[(base + fi) * 3 + 1];
  const float qz = center[(base + fi) * 3 + 2];

  for (int n = tid; n < Nn; n += NT) {
    size_t o = (size_t)(base + n) * 3;
    float dx = center[o + 0] - qx;
    float dy = center[o + 1] - qy;
    float dz = center[o + 2] - qz;
    sd[n] = dx * dx + dy * dy + dz * dz;
  }
  __syncthreads();

  for (int kk = 0; kk < Kk; ++kk) {
    float bd = 3.4e38f;
    int   bi = 0;
    for (int n = tid; n < Nn; n += NT) {
      float d = sd[n];
      if (d < bd) { bd = d; bi = n; }
    }
    rd[tid] = bd;
    ri[tid] = bi;
    __syncthreads();
    for (int o = NT / 2; o > 0; o >>= 1) {
      if (tid < o) {
        float od = rd[tid + o];
        int   oi = ri[tid + o];
        if (od < rd[tid] || (od == rd[tid] && oi < ri[tid])) {
          rd[tid] = od;
          ri[tid] = oi;
        }
      }
      __syncthreads();
    }
    if (tid == 0) {
      knn_idx[((size_t)(b * Ss + s)) * Kk + kk] = ri[0];
      sd[ri[0]] = 3.4e38f;
    }
    __syncthreads();
  }
}

// ---------------------------------------------------------------------------
// 3. Build channel-major input X[74][P] (row 73 is zero padding for WMMA K):
//    0..5  : dx,dy,dz,rho,theta,phi   6..8 : normal   9..72 : feature
// ---------------------------------------------------------------------------
__global__ void build_kernel(const float* __restrict__ center,
                             const float* __restrict__ normal,
                             const float* __restrict__ feature,
                             const int* __restrict__ fps_idx,
                             const int* __restrict__ knn_idx,
                             float* __restrict__ X) {
  const int p = blockIdx.x * blockDim.x + threadIdx.x;
  if (p >= P) return;
  const int s = (p >> 5) & (Ss - 1);
  const int b = p >> 15;

  const int n  = knn_idx[p];
  const int fi = fps_idx[b * Ss + s];
  const size_t base = (size_t)b * Nn;

  float dx = center[(base + n) * 3 + 0] - center[(base + fi) * 3 + 0];
  float dy = center[(base + n) * 3 + 1] - center[(base + fi) * 3 + 1];
  float dz = center[(base + n) * 3 + 2] - center[(base + fi) * 3 + 2];

  const float eps = 1e-10f;
  float rho = sqrtf(dx * dx + dy * dy + dz * dz + eps);
  float zdiv = dz / fmaxf(rho, eps);
  zdiv = fminf(fmaxf(zdiv, -1.0f + eps), 1.0f - eps);
  float theta = acosf(zdiv) / PI_F;
  float xx = dx, yy = dy;
  if (fabsf(xx) < eps) xx += eps;
  if (fabsf(yy) < eps) yy += eps;
  float phi = atan2f(yy, xx) / (2.0f * PI_F) + 0.5f;

  X[(size_t)0 * P + p] = dx;
  X[(size_t)1 * P + p] = dy;
  X[(size_t)2 * P + p] = dz;
  X[(size_t)3 * P + p] = rho;
  X[(size_t)4 * P + p] = theta;
  X[(size_t)5 * P + p] = phi;
  X[(size_t)6 * P + p] = normal[(base + n) * 3 + 0];
  X[(size_t)7 * P + p] = normal[(base + n) * 3 + 1];
  X[(size_t)8 * P + p] = normal[(base + n) * 3 + 2];
  const float* fsrc = feature + (size_t)b * CF * Nn;
  for (int c = 0; c < CF; ++c) {
    X[(size_t)(9 + c) * P + p] = fsrc[(size_t)c * Nn + n];
  }
}

// ---------------------------------------------------------------------------
// Weight zero-padding: Wp[M][CinPad] <- W[M][Cin] (zeros beyond Cin)
// ---------------------------------------------------------------------------
__global__ void pad_w_kernel(const float* __restrict__ W, float* __restrict__ Wp,
                             int M, int Cin, int CinPad) {
  const int i = blockIdx.x * blockDim.x + threadIdx.x;
  if (i >= M * CinPad) return;
  const int m = i / CinPad;
  const int k = i % CinPad;
  Wp[i] = (k < Cin) ? W[m * Cin + k] : 0.f;
}

__global__ void zero_kernel(float* __restrict__ p, int n) {
  const int stride = gridDim.x * blockDim.x;
  for (int i = blockIdx.x * blockDim.x + threadIdx.x; i < n; i += stride)
    p[i] = 0.f;
}

// ---------------------------------------------------------------------------
// 4. GEMM via V_WMMA_F32_16X16X4_F32, guard-free (CIN is padded, compile-time).
//    Y[M][P] = Wpad[M][CIN] * X[CIN][P] + bias
//    Each wave computes a 32(M) x 16(P) tile: two WMMAs per k-step sharing B.
//    grid = (P/64, M/32), block = 128 (4 waves).
// ---------------------------------------------------------------------------
template <int CIN>
__global__ void gemm_wmma_kernel(const float* __restrict__ W,
                                 const float* __restrict__ X,
                                 const float* __restrict__ bias,
                                 float* __restrict__ Y) {
  static_assert(CIN % 4 == 0, "CIN must be padded to multiple of 4");
  const int lane = threadIdx.x & 31;
  const int wave = threadIdx.x >> 5;
  const int half = lane >> 4;
  const int l16  = lane & 15;
  const int pBase = (blockIdx.x * 4 + wave) * 16;
  const int mBase = blockIdx.y * 32;
  const int p = pBase + l16;

  const float* Wr0 = W + (size_t)(mBase + l16) * CIN;        // rows mBase..+15
  const float* Wr1 = W + (size_t)(mBase + 16 + l16) * CIN;   // rows mBase+16..+31
  const float* Xc  = X + p;

  v8f acc0 = {0.f, 0.f, 0.f, 0.f, 0.f, 0.f, 0.f, 0.f};
  v8f acc1 = {0.f, 0.f, 0.f, 0.f, 0.f, 0.f, 0.f, 0.f};
#pragma unroll
  for (int t = 0; t < CIN / 4; ++t) {
    const int k0 = t * 4 + half * 2;
    v2f a0 = *(const v2f*)(Wr0 + k0);          // global_load_b64
    v2f a1 = *(const v2f*)(Wr1 + k0);
    v2f b;
    b.x = Xc[(size_t)k0 * P];
    b.y = Xc[(size_t)(k0 + 1) * P];
    acc0 = wmma4(a0, b, acc0);
    acc1 = wmma4(a1, b, acc1);
  }
#pragma unroll
  for (int v = 0; v < 8; ++v) {
    const int mm0 = mBase + v + 8 * half;
    const int mm1 = mm0 + 16;
    Y[(size_t)mm0 * P + p] = acc0[v] + bias[mm0];
    Y[(size_t)mm1 * P + p] = acc1[v] + bias[mm1];
  }
}

// ---------------------------------------------------------------------------
// 5. Per-channel sum / sumsq (BN stats). grid = (64 chunks, C channels).
// ---------------------------------------------------------------------------
__global__ void stats_kernel(const float* __restrict__ Y,
                             float* __restrict__ st) {
  __shared__ float as[256], as2[256];
  const int c = blockIdx.y;
  const int chunk = P / gridDim.x;
  const int start = blockIdx.x * chunk;
  const int tid = threadIdx.x;
  float s = 0.f, s2 = 0.f;
  for (int i = start + tid; i < start + chunk; i += blockDim.x) {
    float v = Y[(size_t)c * P + i];
    s += v;
    s2 += v * v;
  }
  as[tid] = s;
  as2[tid] = s2;
  __syncthreads();
  for (int o = 128; o > 0; o >>= 1) {
    if (tid < o) { as[tid] += as[tid + o]; as2[tid] += as2[tid + o]; }
    __syncthreads();
  }
  if (tid == 0) {
    atomicAdd(&st[c * 2 + 0], as[0]);
    atomicAdd(&st[c * 2 + 1], as2[0]);
  }
}

// ---------------------------------------------------------------------------
// 6. Layer-0 merge: X = relu(BN(Yloc) + BN(Yfeat))   (64 channels, in-place ok)
// ---------------------------------------------------------------------------
__global__ void norm0_kernel(const float* __restrict__ Yl,
                             const float* __restrict__ Yf,
                             const float* __restrict__ stl,
                             const float* __restrict__ stf,
                             const float* __restrict__ gl, const float* __restrict__ bl,
                             const float* __restrict__ gf, const float* __restrict__ bf,
                             float* __restrict__ X) {
  const size_t i = (size_t)blockIdx.x * blockDim.x + threadIdx.x;
  const int c = (int)(i >> 17);          // / P
  const float invP = 1.0f / (float)P;
  float ml = stl[c * 2] * invP;
  float vl = stl[c * 2 + 1] * invP - ml * ml;
  float mf = stf[c * 2] * invP;
  float vf = stf[c * 2 + 1] * invP - mf * mf;
  float a = gl[c] * (Yl[i] - ml) * rsqrtf(vl + BN_EPS) + bl[c];
  float d = gf[c] * (Yf[i] - mf) * rsqrtf(vf + BN_EPS) + bf[c];
  X[i] = fmaxf(a + d, 0.f);
}

// Generic BN + ReLU, in-place capable.
__global__ void normr_kernel(const float* __restrict__ Y,
                             const float* __restrict__ st,
                             const float* __restrict__ g,
                             const float* __restrict__ be,
                             float* __restrict__ X) {
  const size_t i = (size_t)blockIdx.x * blockDim.x + threadIdx.x;
  const int c = (int)(i >> 17);
  const float invP = 1.0f / (float)P;
  float m = st[c * 2] * invP;
  float v = st[c * 2 + 1] * invP - m * m;
  X[i] = fmaxf(g[c] * (Y[i] - m) * rsqrtf(v + BN_EPS) + be[c], 0.f);
}

// ---------------------------------------------------------------------------
// 7. Fused BN + ReLU + max over K for layer 2 -> Xm[PS][128]
// ---------------------------------------------------------------------------
__global__ void maxbn_kernel(const float* __restrict__ Y,
                             const float* __restrict__ st,
                             const float* __restrict__ g,
                             const float* __restrict__ be,
                             float* __restrict__ Xm) {
  const int i = blockIdx.x * blockDim.x + threadIdx.x;   // 128 * PS threads
  const int c  = i / PS;
  const int ps = i % PS;
  const float invP = 1.0f / (float)P;
  float m = st[c * 2] * invP;
  float v = st[c * 2 + 1] * invP - m * m;
  float is = rsqrtf(v + BN_EPS);
  float gg = g[c], bb = be[c];
  const size_t base = (size_t)c * P + (size_t)ps * Kk;
  float mx = -3.4e38f;
#pragma unroll 4
  for (int k = 0; k < Kk; ++k) {
    float val = fmaxf(gg * (Y[base + k] - m) * is + bb, 0.f);
    mx = fmaxf(mx, val);
  }
  Xm[(size_t)ps * 128 + c] = mx;
}

// ---------------------------------------------------------------------------
// 8. Final projection: Out[PS][256] = Xm[PS][128] * Wp^T + bp  (WMMA, K=128)
// ---------------------------------------------------------------------------
__global__ void final_gemm_kernel(const float* __restrict__ Xm,
                                  const float* __restrict__ Wp,
                                  const float* __restrict__ bp,
                                  float* __restrict__ Out) {
  const int lane = threadIdx.x & 31;
  const int wave = threadIdx.x >> 5;
  const int half = lane >> 4;
  const int l16  = lane & 15;
  const int nBase = (blockIdx.x * 4 + wave) * 16;   // 0..255
  const int mBase = blockIdx.y * 16;                // 0..4095
  const float* Ar = Xm + (size_t)(mBase + l16) * 128;
  const float* Br = Wp + (size_t)(nBase + l16) * 128;
  const int ncol = nBase + l16;

  v8f acc = {0.f, 0.f, 0.f, 0.f, 0.f, 0.f, 0.f, 0.f};
#pragma unroll
  for (int t = 0; t < 32; ++t) {
    const int k0 = t * 4 + half * 2;
    v2f a = *(const v2f*)(Ar + k0);
    v2f b = *(const v2f*)(Br + k0);
    acc = wmma4(a, b, acc);
  }
#pragma unroll
  for (int v = 0; v < 8; ++v) {
    const int mm = mBase + v + 8 * half;
    Out[(size_t)mm * 256 + ncol] = acc[v] + bp[ncol];
  }
}

// ---------------------------------------------------------------------------
// Launch
// ---------------------------------------------------------------------------
extern "C" void kernel_launch(void* const* d_in, const int* in_sizes, int n_in,
                              void* d_out, int out_size, void* d_ws, size_t ws_size,
                              hipStream_t stream) {
  (void)in_sizes; (void)n_in; (void)out_size; (void)ws_size;
  const float* center  = (const float*)d_in[0];
  const float* normal  = (const float*)d_in[1];
  const float* feature = (const float*)d_in[2];
  const float* W_l0 = (const float*)d_in[3];
  const float* b_l0 = (const float*)d_in[4];
  const float* g_l0 = (const float*)d_in[5];
  const float* be_l0 = (const float*)d_in[6];
  const float* W_f0 = (const float*)d_in[7];
  const float* b_f0 = (const float*)d_in[8];
  const float* g_f0 = (const float*)d_in[9];
  const float* be_f0 = (const float*)d_in[10];
  const float* W1 = (const float*)d_in[11];
  const float* b1 = (const float*)d_in[12];
  const float* g1 = (const float*)d_in[13];
  const float* be1 = (const float*)d_in[14];
  const float* W2 = (const float*)d_in[15];
  const float* b2 = (const float*)d_in[16];
  const float* g2 = (const float*)d_in[17];
  const float* be2 = (const float*)d_in[18];
  const float* Wp = (const float*)d_in[19];
  const float* bp = (const float*)d_in[20];

  // workspace layout (all region sizes even -> 8B alignment preserved)
  char* ws = (char*)d_ws;
  int* fps_idx = (int*)ws;                                   // B*S
  int* knn_idx = fps_idx + Bb * Ss;                          // B*S*K
  float* stats = (float*)(knn_idx + P);                      // 1024 floats
  float* Wl0p  = stats + 1024;                               // 64*8
  float* Wf0p  = Wl0p + 64 * 8;                              // 64*68
  float* Xin   = Wf0p + 64 * 68;                             // 74*P (row 73 zero)
  float* Y0loc = Xin + (size_t)74 * P;                       // 64*P (reused as X1)
  float* Y0feat= Y0loc + (size_t)64 * P;                     // 64*P
  float* Y1    = Y0feat + (size_t)64 * P;                    // 64*P (reused as X2)
  float* Y2    = Y1 + (size_t)64 * P;                        // 128*P
  float* Xm    = Y2 + (size_t)128 * P;                       // PS*128

  float* out_center = (float*)d_out;
  float* out_normal = out_center + (size_t)Bb * Ss * 3;
  float* out_final  = out_normal + (size_t)Bb * Ss * 3;

  float* st_loc  = stats + 0;     // 64*2
  float* st_feat = stats + 128;   // 64*2
  float* st_1    = stats + 256;   // 64*2
  float* st_2    = stats + 384;   // 128*2

  zero_kernel<<<1, 256, 0, stream>>>(stats, 1024);
  zero_kernel<<<64, 256, 0, stream>>>(Xin + (size_t)73 * P, P);  // WMMA K pad row
  pad_w_kernel<<<(64 * 8 + 255) / 256, 256, 0, stream>>>(W_l0, Wl0p, 64, 6, 8);
  pad_w_kernel<<<(64 * 68 + 255) / 256, 256, 0, stream>>>(W_f0, Wf0p, 64, 67, 68);

  fps_kernel<<<Bb, 1024, 0, stream>>>(center, normal, fps_idx, out_center, out_normal);
  knn_kernel<<<dim3(Ss, Bb), 256, 0, stream>>>(center, fps_idx, knn_idx);
  build_kernel<<<P / 256, 256, 0, stream>>>(center, normal, feature, fps_idx, knn_idx, Xin);

  // layer 0 (split pos / feat branches)
  gemm_wmma_kernel<8><<<dim3(P / 64, 2), 128, 0, stream>>>(Wl0p, Xin, b_l0, Y0loc);
  gemm_wmma_kernel<68><<<dim3(P / 64, 2), 128, 0, stream>>>(Wf0p, Xin + (size_t)6 * P, b_f0, Y0feat);
  stats_kernel<<<dim3(64, 64), 256, 0, stream>>>(Y0loc, st_loc);
  stats_kernel<<<dim3(64, 64), 256, 0, stream>>>(Y0feat, st_feat);
  norm0_kernel<<<(64 * P) / 256, 256, 0, stream>>>(Y0loc, Y0feat, st_loc, st_feat,
                                                   g_l0, be_l0, g_f0, be_f0, Y0loc);

  // layer 1
  gemm_wmma_kernel<64><<<dim3(P / 64, 2), 128, 0, stream>>>(W1, Y0loc, b1, Y1);
  stats_kernel<<<dim3(64, 64), 256, 0, stream>>>(Y1, st_1);
  normr_kernel<<<(64 * P) / 256, 256, 0, stream>>>(Y1, st_1, g1, be1, Y1);

  // layer 2
  gemm_wmma_kernel<64><<<dim3(P / 64, 4), 128, 0, stream>>>(W2, Y1, b2, Y2);
  stats_kernel<<<dim3(64, 128), 256, 0, stream>>>(Y2, st_2);
  maxbn_kernel<<<(128 * PS) / 256, 256, 0, stream>>>(Y2, st_2, g2, be2, Xm);

  // final projection
  final_gemm_kernel<<<dim3(4, PS / 16), 128, 0, stream>>>(Xm, Wp, bp, out_final);
}